// DenseCapsule_15642270892808
// MI455X (gfx1250) — compile-verified
//
#include <hip/hip_runtime.h>

typedef __attribute__((ext_vector_type(2))) float v2f;
typedef __attribute__((ext_vector_type(8))) float v8f;

#define B_TOT   256
#define O_TOT   10
#define I_TOT   1152
#define DI      8
#define DO      16
#define BT      (B_TOT / 16)            // 16 batch tiles
#define XB_STRIDE (I_TOT * DI)          // 9216 floats per batch row of x
#define COUP_SLICE (B_TOT * O_TOT * I_TOT)  // 2,949,120
#define OUT_ELEMS  (B_TOT * O_TOT * DO)     // 40,960

// ---------------------------------------------------------------------------
// coups[0] = softmax(zeros) = 1/O = 0.1  (also serves as c for iteration 0)
// ---------------------------------------------------------------------------
__global__ void fill_kernel(float* __restrict__ p, int n, float v) {
    int t = blockIdx.x * blockDim.x + threadIdx.x;
    if (t < n) p[t] = v;
}

// ---------------------------------------------------------------------------
// s[b,o,:] = sum_i c[b,o,i] * x_hat[b,o,i,:]  computed as one long-K fp32 GEMM
//   A[b, (i,m)] = c[b,o,i] * x[b,i,m]   (M=16 batches, K=I*DI=9216)
//   B[(i,m), d] = w[o,i,d,m]            (N=16 capsule dims)
// followed by squash() on the 16-dim result.
// One block per (batch_tile, o); 8 waves split the I range.
// ---------------------------------------------------------------------------
__global__ __launch_bounds__(256)
void wsum_squash_kernel(const float* __restrict__ x,
                        const float* __restrict__ w,
                        const float* __restrict__ c,
                        float* __restrict__ outbuf) {
    __shared__ float red[8 * 256];

    const int blk  = blockIdx.x;
    const int bt   = blk / O_TOT;
    const int o    = blk % O_TOT;
    const int b0   = bt * 16;
    const int tid  = threadIdx.x;
    const int wv   = tid >> 5;          // wave id 0..7
    const int lane = tid & 31;
    const int half = lane >> 4;         // K-pair select per ISA A/B layout
    const int ln16 = lane & 15;         // A row (batch) == B col (d)

    const int i_beg = wv * (I_TOT / 8);
    const int i_end = i_beg + (I_TOT / 8);

    // loop-invariant per-lane base pointers
    const float* cp = c + ((long)(b0 + ln16) * O_TOT + o) * I_TOT;
    const float* xp = x + (long)(b0 + ln16) * XB_STRIDE + (long)i_beg * DI
                        + 2 * half;
    const float* wp = w + (((long)o * I_TOT + i_beg) * DO + ln16) * DI
                        + 2 * half;

    v8f acc = {};
    for (int i = i_beg; i < i_end; ++i) {
        __builtin_prefetch(wp + 8 * DO * DI, 0, 1);   // global_prefetch_b8
        const float cv = cp[i];
        // A operand: c * x[b, i, m]; lanes 0-15 take K={0,1}, 16-31 K={2,3}
        v2f a0 = { cv * xp[0], cv * xp[1] };
        v2f a1 = { cv * xp[4], cv * xp[5] };
        // B operand: w[o, i, d=ln16, m=k] (m contiguous -> aligned pairs)
        v2f bv0 = { wp[0], wp[1] };
        v2f bv1 = { wp[4], wp[5] };
        acc = __builtin_amdgcn_wmma_f32_16x16x4_f32(false, a0, false, bv0,
                                                    (short)0, acc, false, false);
        acc = __builtin_amdgcn_wmma_f32_16x16x4_f32(false, a1, false, bv1,
                                                    (short)0, acc, false, false);
        xp += DI;
        wp += DO * DI;
    }

    // cross-wave reduction of the 16x16 accumulator tile
#pragma unroll
    for (int r = 0; r < 8; ++r)
        red[wv * 256 + r * 32 + lane] = acc[r];
    __syncthreads();
    float s = 0.f;
#pragma unroll
    for (int w8 = 0; w8 < 8; ++w8) s += red[w8 * 256 + tid];

    // thread t holds s[m,n] with r=t>>5, l=t&31 -> m=r+8*(l>>4), n=l&15
    const int r = tid >> 5;
    const int l = tid & 31;
    const int m = r + 8 * (l >> 4);
    const int n = l & 15;

    // squash: norm over the 16 d-values lives in one 16-lane group
    float nsq = s * s;
#pragma unroll
    for (int off = 8; off >= 1; off >>= 1)
        nsq += __shfl_xor(nsq, off, 32);
    const float nrm   = sqrtf(nsq);
    const float scale = nsq / (1.f + nsq) / (nrm + 1e-8f);

    outbuf[((long)(b0 + m) * O_TOT + o) * DO + n] = scale * s;
}

// ---------------------------------------------------------------------------
// dot[b,o,i] = sum_d out[b,o,d] * x_hat[b,o,i,d];  bb (+)= dot
// x_hat tile recomputed transposed: T[d,b] = W(16x8) @ X^T(8x16) via 2 WMMAs,
// so each lane owns one batch column and reduces over d in-register.
// ---------------------------------------------------------------------------
__global__ __launch_bounds__(256)
void bupdate_kernel(const float* __restrict__ x,
                    const float* __restrict__ w,
                    const float* __restrict__ outv,   // [B,O,DO]
                    float* __restrict__ bb,           // [B,O,I]
                    int accum) {
    const int blk  = blockIdx.x;
    const int bt   = blk / O_TOT;
    const int o    = blk % O_TOT;
    const int b0   = bt * 16;
    const int tid  = threadIdx.x;
    const int wv   = tid >> 5;
    const int lane = tid & 31;
    const int half = lane >> 4;
    const int n    = lane & 15;   // A row = d index; B col = batch index

    // preload this lane's 8 out values: batch = b0+n, d = r + 8*half
    float oreg[8];
    const long obase = ((long)(b0 + n) * O_TOT + o) * DO + 8 * half;
#pragma unroll
    for (int r = 0; r < 8; ++r) oreg[r] = outv[obase + r];

    const int i_beg = wv * (I_TOT / 8);
    const int i_end = i_beg + (I_TOT / 8);

    const float* wp = w + (((long)o * I_TOT + i_beg) * DO + n) * DI + 2 * half;
    const float* xp = x + (long)(b0 + n) * XB_STRIDE + (long)i_beg * DI
                        + 2 * half;
    float* bp = bb + ((long)(b0 + n) * O_TOT + o) * I_TOT;

    for (int i = i_beg; i < i_end; ++i) {
        __builtin_prefetch(wp + 8 * DO * DI, 0, 1);
        v2f a0 = { wp[0], wp[1] };
        v2f a1 = { wp[4], wp[5] };
        v2f bv0 = { xp[0], xp[1] };
        v2f bv1 = { xp[4], xp[5] };
        v8f t = {};
        t = __builtin_amdgcn_wmma_f32_16x16x4_f32(false, a0, false, bv0,
                                                  (short)0, t, false, false);
        t = __builtin_amdgcn_wmma_f32_16x16x4_f32(false, a1, false, bv1,
                                                  (short)0, t, false, false);
        float acc = 0.f;
#pragma unroll
        for (int r = 0; r < 8; ++r) acc += t[r] * oreg[r];
        acc += __shfl_xor(acc, 16, 32);    // fold d halves across lane halves
        if (half == 0)
            bp[i] = accum ? (bp[i] + acc) : acc;
        wp += DO * DI;
        xp += DI;
    }
}

// ---------------------------------------------------------------------------
// c[b,o,i] = 0.01 + 0.99*(bb - min_o bb)/(max_o bb - min_o bb)
// ---------------------------------------------------------------------------
__global__ void coup_kernel(const float* __restrict__ bb,
                            float* __restrict__ cdst) {
    const int t = blockIdx.x * blockDim.x + threadIdx.x;
    if (t >= B_TOT * I_TOT) return;
    const int b = t / I_TOT;
    const int i = t % I_TOT;
    float v[O_TOT];
    float mn = 1e30f, mx = -1e30f;
#pragma unroll
    for (int o = 0; o < O_TOT; ++o) {
        v[o] = bb[((long)b * O_TOT + o) * I_TOT + i];
        mn = fminf(mn, v[o]);
        mx = fmaxf(mx, v[o]);
    }
    const float inv = 0.99f / (mx - mn);
#pragma unroll
    for (int o = 0; o < O_TOT; ++o)
        cdst[((long)b * O_TOT + o) * I_TOT + i] = 0.01f + (v[o] - mn) * inv;
}

// ---------------------------------------------------------------------------
extern "C" void kernel_launch(void* const* d_in, const int* in_sizes, int n_in,
                              void* d_out, int out_size, void* d_ws, size_t ws_size,
                              hipStream_t stream) {
    (void)in_sizes; (void)n_in; (void)out_size; (void)ws_size;
    const float* x = (const float*)d_in[0];   // [256, 1152, 8]
    const float* w = (const float*)d_in[1];   // [10, 1152, 16, 8]

    float* out   = (float*)d_out;             // [256,10,16]
    float* coups = out + OUT_ELEMS;           // [3,256,10,1152]
    float* c0 = coups;
    float* c1 = coups + (long)COUP_SLICE;
    float* c2 = coups + 2L * COUP_SLICE;

    float* ws     = (float*)d_ws;
    float* outbuf = ws;                       // [256,10,16] scratch
    float* bb     = ws + OUT_ELEMS;           // [256,10,1152] routing logits

    const dim3 blk(256);
    const int gemm_blocks = BT * O_TOT;       // 160
    const int coup_blocks = (B_TOT * I_TOT + 255) / 256;

    // coups[0] = uniform 1/O; doubles as the iteration-0 coupling buffer,
    // eliminating the nullptr branch from the GEMM inner loop.
    fill_kernel<<<(COUP_SLICE + 255) / 256, blk, 0, stream>>>(c0, COUP_SLICE, 0.1f);

    // routing iteration 0
    wsum_squash_kernel<<<gemm_blocks, blk, 0, stream>>>(x, w, c0, outbuf);
    bupdate_kernel<<<gemm_blocks, blk, 0, stream>>>(x, w, outbuf, bb, 0);
    coup_kernel<<<coup_blocks, blk, 0, stream>>>(bb, c1);

    // routing iteration 1
    wsum_squash_kernel<<<gemm_blocks, blk, 0, stream>>>(x, w, c1, outbuf);
    bupdate_kernel<<<gemm_blocks, blk, 0, stream>>>(x, w, outbuf, bb, 1);
    coup_kernel<<<coup_blocks, blk, 0, stream>>>(bb, c2);

    // routing iteration 2 (final): write squashed result straight to d_out
    wsum_squash_kernel<<<gemm_blocks, blk, 0, stream>>>(x, w, c2, out);
}